// DicRBF_78314433675988
// MI455X (gfx1250) — compile-verified
//
#include <hip/hip_runtime.h>
#include <hip/hip_bf16.h>

typedef __attribute__((ext_vector_type(2))) float v2f;
typedef __attribute__((ext_vector_type(4))) float v4f;
typedef __attribute__((ext_vector_type(8))) float v8f;

#define NROWS      131072
#define DDIM       64
#define KCENT      512
#define M_TILE     128
#define THREADS    256
#define LDS_STRIDE 68      // 64 + 4 pad: bank = 4*row + col (mod 64) -> conflict-free
#define OUT_STRIDE 577     // 1 + 64 + 512
#define REGZ       1e-4f
#define LN2        0.69314718055994531f

__global__ __launch_bounds__(THREADS)
void DicRBF_78314433675988_kernel(const float* __restrict__ data,
                                  const float* __restrict__ centers,
                                  float* __restrict__ out)
{
    extern __shared__ float lds[];                 // [512*68] centers + [512] norms
    float* cen   = lds;
    float* cnorm = lds + KCENT * LDS_STRIDE;

    const int tid  = threadIdx.x;
    const int lane = tid & 31;
    const int wave = tid >> 5;
    const int h    = lane >> 4;     // which half of the wave (K split)
    const int nn   = lane & 15;     // N / M index within tile
    const int block_row = blockIdx.x * M_TILE;

    // ---- cooperative load of all 512 centers into padded LDS (b128 loads) ----
    for (int f = tid; f < KCENT * DDIM / 4; f += THREADS) {
        v4f v = *(const v4f*)(centers + 4 * f);
        int row = (4 * f) / DDIM;
        int col = (4 * f) % DDIM;
        *(v4f*)(cen + row * LDS_STRIDE + col) = v;
    }
    __syncthreads();

    // ---- center squared norms into LDS ----
    for (int c = tid; c < KCENT; c += THREADS) {
        const float* cr = cen + c * LDS_STRIDE;
        float s = 0.f;
        #pragma unroll
        for (int k = 0; k < DDIM; ++k) s += cr[k] * cr[k];
        cnorm[c] = s;
    }

    // ---- ones column + data passthrough (streamed, non-temporal) ----
    for (int i = tid; i < M_TILE * (1 + DDIM); i += THREADS) {
        int row = block_row + i / (1 + DDIM);
        int col = i % (1 + DDIM);
        float v = (col == 0) ? 1.0f : data[(size_t)row * DDIM + (col - 1)];
        __builtin_nontemporal_store(v, out + (size_t)row * OUT_STRIDE + col);
    }
    __syncthreads();

    // ---- per-wave A fragment: 16 rows x 64 K, held in registers for the whole sweep ----
    // ISA layout 32-bit A 16x4: lanes 0-15 hold K=0,1 ; lanes 16-31 hold K=2,3
    const int m_base = block_row + wave * 16;
    const int my_row = m_base + nn;
    v2f a[16];
    #pragma unroll
    for (int j = 0; j < 16; ++j)
        a[j] = *(const v2f*)(data + (size_t)my_row * DDIM + 4 * j + 2 * h);

    // row squared norms: lane holds half its row; combine halves via shfl_xor(16)
    float part = 0.f;
    #pragma unroll
    for (int j = 0; j < 16; ++j) part += a[j].x * a[j].x + a[j].y * a[j].y;
    float full = part + __shfl_xor(part, 16, 32);
    // accumulator element (vgpr j, lane l) is row M = j + 8*(l/16)
    float xnv[8];
    #pragma unroll
    for (int j = 0; j < 8; ++j) xnv[j] = __shfl(full, j + 8 * h, 32);

    // ---- sweep 512 centers in 16-wide tiles: 16 x v_wmma_f32_16x16x4_f32 each ----
    for (int cb = 0; cb < KCENT; cb += 16) {
        v8f acc = {};
        // B[k][n] = centers[cb+n][k]; lane n reads its center row from LDS
        const float* brow = cen + (cb + nn) * LDS_STRIDE + 2 * h;
        #pragma unroll
        for (int j = 0; j < 16; ++j) {
            v2f b = *(const v2f*)(brow + 4 * j);
            acc = __builtin_amdgcn_wmma_f32_16x16x4_f32(
                      /*neg_a=*/false, a[j], /*neg_b=*/false, b,
                      /*c_mod=*/(short)0, acc, /*reuse_a=*/false, /*reuse_b=*/false);
        }
        float cn = cnorm[cb + nn];
        // Branchless epilogue: with the 1e-4 regularizer, log(r+reg) is always
        // finite, and d2==0 => 0 * finite = 0, matching xlogy semantics exactly.
        #pragma unroll
        for (int j = 0; j < 8; ++j) {
            float d2 = xnv[j] + cn - 2.0f * acc[j];
            d2 = fmaxf(d2, 0.0f);
            float r = __builtin_amdgcn_sqrtf(d2);            // raw v_sqrt_f32
            float lg = __builtin_amdgcn_logf(r + REGZ) * LN2; // v_log_f32 (log2) * ln2
            float rbf = d2 * lg;
            int row = m_base + j + 8 * h;
            __builtin_nontemporal_store(
                rbf, out + (size_t)row * OUT_STRIDE + 1 + DDIM + cb + nn);
        }
    }
}

extern "C" void kernel_launch(void* const* d_in, const int* in_sizes, int n_in,
                              void* d_out, int out_size, void* d_ws, size_t ws_size,
                              hipStream_t stream) {
    const float* data    = (const float*)d_in[0];   // [131072, 64]
    const float* centers = (const float*)d_in[1];   // [512, 64]
    float* out = (float*)d_out;                     // [131072, 577]
    (void)in_sizes; (void)n_in; (void)out_size; (void)d_ws; (void)ws_size;

    size_t shmem = (size_t)(KCENT * LDS_STRIDE + KCENT) * sizeof(float); // ~141 KB
    dim3 grid(NROWS / M_TILE);   // 1024 blocks
    dim3 block(THREADS);         // 8 waves (wave32)
    DicRBF_78314433675988_kernel<<<grid, block, shmem, stream>>>(data, centers, out);
}